// ErrorRateInfluencedLoss_9878424781112
// MI455X (gfx1250) — compile-verified
//
#include <hip/hip_runtime.h>
#include <float.h>

// ---------------------------------------------------------------------------
// ErrorRateInfluencedLoss for MI455X (gfx1250, wave32)
//   out[0] = CTC_W*ctc + ER_W*soft, out[1] = ctc, out[2] = error_rate
// Pipeline:
//   K1 rowstats : single 128MB streaming pass -> lse[B*T], argmax[B*T]
//                 (per-wave 16 rows; v_wmma_f32_16x16x4_f32 ones-B row-sum)
//   K2 seq      : fused sequential phase, 96 blocks x 288 thr:
//                 blocks 0..31  CTC alpha recurrence (257 states, LDS ping-pong,
//                               ONE barrier/step, register-double-buffered gather)
//                 blocks 32..63 collapse + Levenshtein (async-to-LDS staging,
//                               wave32 cummin-scan DP)
//                 blocks 64..95 masked cross-entropy term
//   K3 finalize : tiny reduction
// ---------------------------------------------------------------------------

#define Bb 32
#define Tt 1024
#define Vv 1024
#define Uu 128
#define Ss 257            // 2*U+1
#define NEGC (-1.0e30f)
#define BIGI 0x3FFFFFF

typedef __attribute__((ext_vector_type(2))) float v2f;
typedef __attribute__((ext_vector_type(8))) float v8f;
typedef __attribute__((ext_vector_type(4))) int   v4i_t;

#if defined(__gfx1250__) && __has_builtin(__builtin_amdgcn_wmma_f32_16x16x4_f32)
#define HAVE_WMMA_F32X4 1
#else
#define HAVE_WMMA_F32X4 0
#endif

#if defined(__gfx1250__) && \
    __has_builtin(__builtin_amdgcn_global_load_async_to_lds_b128) && \
    __has_builtin(__builtin_amdgcn_s_wait_asynccnt)
#define HAVE_ASYNC_LDS 1
#else
#define HAVE_ASYNC_LDS 0
#endif

// ------------------------------- K1: row stats ------------------------------
// 512 blocks x 128 threads (4 waves). Each wave owns 16 consecutive (b,t) rows.
// lane L -> row R0 + (L&15); lanes L and L+16 split the 1024 columns.
// Per 32-element tile: tile max/argmax + tile expsum, then one online rescale.
__global__ __launch_bounds__(128) void rowstats_kernel(
    const float* __restrict__ logits,
    float* __restrict__ lse_out,
    int* __restrict__ best_out) {
  const int lane = threadIdx.x & 31;
  const int wid  = threadIdx.x >> 5;
  const int task = blockIdx.x * 4 + wid;          // 0..2047
  const int R0   = task * 16;
  const int half = lane >> 4;
  const int r    = R0 + (lane & 15);
  const float* __restrict__ row = logits + (size_t)r * Vv;

  float m = -3.0e38f, s = 0.0f;
  int gidx = 0;

  #pragma unroll 1
  for (int ct = 0; ct < 16; ++ct) {
    const int c0 = ct * 64 + half * 32;           // this lane's 32 columns
    const float4* __restrict__ p = (const float4*)(row + c0);
    if (ct + 1 < 16) __builtin_prefetch(row + c0 + 64, 0, 1);  // global_prefetch_b8

    float xv[32];
    #pragma unroll
    for (int k = 0; k < 8; ++k) {
      float4 v = p[k];
      xv[4*k+0] = v.x; xv[4*k+1] = v.y; xv[4*k+2] = v.z; xv[4*k+3] = v.w;
    }
    // tile max + argmax (first occurrence: strict >)
    float tm = xv[0]; int ti = c0;
    #pragma unroll
    for (int e = 1; e < 32; ++e) {
      if (xv[e] > tm) { tm = xv[e]; ti = c0 + e; }
    }
    // tile expsum against tile max
    float ts = 0.0f;
    #pragma unroll
    for (int e = 0; e < 32; ++e) ts += __expf(xv[e] - tm);
    // online merge (one rescale per tile)
    if (tm > m) gidx = ti;                        // strict > keeps earliest index
    float M = fmaxf(m, tm);
    s = s * __expf(m - M) + ts * __expf(tm - M);
    m = M;
  }

  // merge the two half-row lanes (L <-> L^16)
  float om = __shfl_xor(m, 16, 32);
  int   oi = __shfl_xor(gidx, 16, 32);
  int fidx = (om > m) ? oi : ((om < m) ? gidx : ((oi < gidx) ? oi : gidx));
  float Mg = fmaxf(m, om);
  float a0 = s * __expf(m - Mg);                  // this lane's rescaled partial

  float dk[8];
#if HAVE_WMMA_F32X4
  // A 16x4 f32 layout: lane m holds A[m][0..1], lane m+16 holds A[m][2..3].
  // Put partial in K-slot 0, zero in slot 1; B = ones(4x16); C = 0.
  // D[m][n] = partial(lane m) + partial(lane m+16) = full row expsum, all n.
  v2f a;  a[0] = a0;  a[1] = 0.0f;
  v2f bo; bo[0] = 1.0f; bo[1] = 1.0f;
  v8f c = {};
  v8f d = __builtin_amdgcn_wmma_f32_16x16x4_f32(
      /*neg_a=*/false, a, /*neg_b=*/false, bo,
      /*c_mod=*/(short)0, c, /*reuse_a=*/false, /*reuse_b=*/false);
  #pragma unroll
  for (int k = 0; k < 8; ++k) dk[k] = d[k];       // lanes 0-15: rows 0..7; 16-31: rows 8..15
#else
  float rs = a0 + __shfl_xor(a0, 16, 32);         // row sum lives in lanes rho, rho+16
  const int b0f = (lane >= 16) ? 8 : 0;
  #pragma unroll
  for (int k = 0; k < 8; ++k) dk[k] = __shfl(rs, b0f + k, 32);
#endif

  const int base = (lane >= 16) ? 8 : 0;
  float lv[8];
  #pragma unroll
  for (int k = 0; k < 8; ++k) {
    float mr = __shfl(Mg, base + k, 32);          // row (base+k)'s max
    lv[k] = mr + __logf(dk[k]);
  }
  if (lane == 0 || lane == 16) {
    #pragma unroll
    for (int k = 0; k < 8; ++k) lse_out[R0 + base + k] = lv[k];
  }
  if (lane < 16) best_out[R0 + lane] = fidx;
}

// ------------------------- K2: fused sequential phase -----------------------
__global__ __launch_bounds__(288) void seq_kernel(
    const float* __restrict__ logits,
    const int* __restrict__ targets,
    const int* __restrict__ input_lengths,
    const int* __restrict__ target_lengths,
    const float* __restrict__ lse,
    const int* __restrict__ best,
    float* __restrict__ ctc_part,
    float* __restrict__ soft_part,
    int* __restrict__ dist_out,
    int* __restrict__ tlenc_out) {
  __shared__ float bufA[Ss], bufB[Ss];
  __shared__ int tgt[Uu];
  __shared__ __align__(16) int bestrow[Tt];
  __shared__ int pred[Tt];
  __shared__ int tcol[Uu];
  __shared__ int plen_s, tlen_s;
  __shared__ float sred[8];

  const int role = blockIdx.x >> 5;               // 0=CTC, 1=ED, 2=soft
  const int b    = blockIdx.x & 31;
  const int tid  = threadIdx.x;

  if (role == 0) {
    // ---------------- CTC alpha recurrence ----------------
    if (tid < Uu) tgt[tid] = targets[b * Uu + tid];
    __syncthreads();
    const int il = input_lengths[b];
    const int tl = target_lengths[b];
    const int s  = tid;
    const bool act = (s < Ss);
    int lab = 0; bool skip = false;
    if (act && (s & 1)) {
      lab = tgt[s >> 1];
      if (s >= 3) skip = (lab != tgt[(s >> 1) - 1]);
    }
    const float* __restrict__ lrow = logits + (size_t)b * Tt * Vv;
    const float* __restrict__ gp   = lrow + lab;  // gather pointer, stride Vv per step
    const float* __restrict__ lseb = lse + b * Tt;

    // t = 0 init
    if (act) {
      float v = NEGC;
      if (s == 0) v = lrow[0] - lseb[0];          // blank
      else if (s == 1) v = lrow[lab] - lseb[0];
      bufA[s] = v;
    }
    __syncthreads();

    // register double-buffer for the label gather: load t+1 before computing t,
    // so the L2 latency overlaps the logsumexp chain and the barrier.
    float gv = act ? gp[(size_t)Vv] : 0.0f;       // value for t = 1
    float* cur = bufA; float* nxt = bufB;
    for (int t = 1; t < Tt; ++t) {
      float gv_next = 0.0f;
      if (act && (t + 1 < Tt)) gv_next = gp[(size_t)(t + 1) * Vv];
      const float lset = lseb[t];
      float outv = 0.0f;
      if (act) {
        float lp = gv - lset;
        float a  = cur[s];
        float s1 = (s >= 1) ? cur[s - 1] : NEGC;
        float s2 = skip ? cur[s - 2] : NEGC;
        float mm = fmaxf(a, fmaxf(s1, s2));
        float nv = mm + __logf(__expf(a - mm) + __expf(s1 - mm) + __expf(s2 - mm)) + lp;
        outv = (t < il) ? nv : a;
        nxt[s] = outv;                            // nxt != cur: no WAR hazard
      }
      __syncthreads();                            // single barrier per step
      float* tp = cur; cur = nxt; nxt = tp;
      gv = gv_next;
    }
    if (tid == 0) {
      int last = 2 * tl;
      float v1 = cur[last];
      float v2 = cur[(last - 1) > 0 ? (last - 1) : 0];
      float mm = fmaxf(v1, v2);
      float ll = mm + __logf(__expf(v1 - mm) + __expf(v2 - mm));
      float loss = -ll;
      if (loss > 1e29f) loss = 0.0f;              // zero_infinity
      ctc_part[b] = loss / (float)tl;
    }
  } else if (role == 1) {
    // ---------------- collapse + edit distance ----------------
    // Cooperatively stage the 4KB argmax row into LDS: 256 lanes x B128 async.
#if HAVE_ASYNC_LDS
    if (tid < 256) {
      __builtin_amdgcn_global_load_async_to_lds_b128(
          (__attribute__((address_space(1))) v4i_t*)(best + (size_t)b * Tt + tid * 4),
          (__attribute__((address_space(3))) v4i_t*)(&bestrow[tid * 4]),
          0, 0);
    }
    __builtin_amdgcn_s_wait_asynccnt(0);
#else
    if (tid < 256) {
      ((int4*)bestrow)[tid] = ((const int4*)(best + (size_t)b * Tt))[tid];
    }
#endif
    __syncthreads();
    if (tid == 0) {
      int il = input_lengths[b];
      int prev = -1, n = 0;
      for (int t = 0; t < il; ++t) {
        int c = bestrow[t];
        if (c != 0 && c != prev) pred[n++] = c;
        prev = c;
      }
      plen_s = n;
      int tl = target_lengths[b];
      prev = -1; n = 0;
      for (int u = 0; u < tl; ++u) {
        int c = targets[b * Uu + u];
        if (c != 0 && c != prev) tcol[n++] = c;
        prev = c;
      }
      tlen_s = n;
      tlenc_out[b] = n;
    }
    __syncthreads();
    if (tid < 32) {
      const int lane = tid;
      const int plen = plen_s, tlen = tlen_s;     // tlen <= 128 -> cols j=0..tlen (<=129)
      int P[5];                                    // lane owns j = 5*lane + k
      #pragma unroll
      for (int k = 0; k < 5; ++k) P[k] = 5 * lane + k;   // dp row 0: P[j] = j
      for (int i = 1; i <= plen; ++i) {
        int ai = pred[i - 1];
        int pleft = __shfl_up(P[4], 1, 32);       // P[j-1] across lane boundary
        int v[5];
        #pragma unroll
        for (int k = 0; k < 5; ++k) {
          int j = 5 * lane + k;
          int bv;
          if (j == 0) bv = i;
          else if (j <= tlen) {
            int pjm1 = (k == 0) ? pleft : P[k - 1];
            int cost = (ai != tcol[j - 1]) ? 1 : 0;
            int t1 = P[k] + 1, t2 = pjm1 + cost;
            bv = (t1 < t2) ? t1 : t2;
          } else bv = BIGI;
          v[k] = bv - j;
        }
        // wave-wide inclusive cummin of v, then new[j] = j + cummin
        int loc = v[0];
        #pragma unroll
        for (int k = 1; k < 5; ++k) loc = (v[k] < loc) ? v[k] : loc;  // lane total
        int incl = loc;
        #pragma unroll
        for (int off = 1; off < 32; off <<= 1) {
          int x = __shfl_up(incl, off, 32);
          if (lane >= off && x < incl) incl = x;
        }
        int excl = __shfl_up(incl, 1, 32);
        if (lane == 0) excl = BIGI;
        int run = excl;
        #pragma unroll
        for (int k = 0; k < 5; ++k) {
          run = (v[k] < run) ? v[k] : run;
          P[k] = 5 * lane + k + run;
        }
      }
      #pragma unroll
      for (int k = 0; k < 5; ++k)
        if (5 * lane + k == tlen) dist_out[b] = P[k];
    }
  } else {
    // ---------------- masked cross-entropy term ----------------
    const int il = input_lengths[b];
    const int tl = target_lengths[b];
    const int ml = (il < tl) ? il : tl;
    float nll = 0.0f;
    if (tid < Uu && tid < ml) {
      int tg = targets[b * Uu + tid];
      nll = lse[b * Tt + tid] - logits[((size_t)b * Tt + tid) * Vv + tg];
    }
    if (tid < 128) {
      const int lane = tid & 31, w = tid >> 5;
      #pragma unroll
      for (int off = 16; off > 0; off >>= 1) nll += __shfl_down(nll, off, 32);
      if (lane == 0) sred[w] = nll;
    }
    __syncthreads();
    if (tid == 0) {
      float tot = sred[0] + sred[1] + sred[2] + sred[3];
      soft_part[b] = tot / (float)(ml > 0 ? ml : 1);
    }
  }
}

// ------------------------------- K3: finalize -------------------------------
__global__ __launch_bounds__(32) void finalize_kernel(
    const float* __restrict__ ctc_part,
    const float* __restrict__ soft_part,
    const int* __restrict__ dist,
    const int* __restrict__ tlenc,
    float* __restrict__ out) {
  const int lane = threadIdx.x;
  float c  = ctc_part[lane];
  float sp = soft_part[lane];
  int   dd = dist[lane];
  int   tk = tlenc[lane];
  #pragma unroll
  for (int off = 16; off > 0; off >>= 1) {
    c  += __shfl_down(c, off, 32);
    sp += __shfl_down(sp, off, 32);
    dd += __shfl_down(dd, off, 32);
    tk += __shfl_down(tk, off, 32);
  }
  if (lane == 0) {
    float ctc  = c / (float)Bb;
    float soft = sp / (float)Bb;
    float er   = (tk == 0) ? 0.0f : ((float)dd / (float)tk);
    out[0] = 1.0f * ctc + 0.5f * soft;
    out[1] = ctc;
    out[2] = er;
  }
}

// ---------------------------------------------------------------------------
extern "C" void kernel_launch(void* const* d_in, const int* in_sizes, int n_in,
                              void* d_out, int out_size, void* d_ws, size_t ws_size,
                              hipStream_t stream) {
  const float* logits         = (const float*)d_in[0];
  const int*   targets        = (const int*)d_in[1];
  const int*   input_lengths  = (const int*)d_in[2];
  const int*   target_lengths = (const int*)d_in[3];
  float* out = (float*)d_out;

  // workspace layout
  float* lse      = (float*)d_ws;                 // B*T
  int*   best     = (int*)(lse + Bb * Tt);        // B*T
  float* ctc_part = (float*)(best + Bb * Tt);     // B
  float* soft_part= ctc_part + Bb;                // B
  int*   dist     = (int*)(soft_part + Bb);       // B
  int*   tlenc    = dist + Bb;                    // B

  (void)in_sizes; (void)n_in; (void)out_size; (void)ws_size;

  rowstats_kernel<<<(Bb * Tt) / (16 * 4), 128, 0, stream>>>(logits, lse, best);
  seq_kernel<<<96, 288, 0, stream>>>(logits, targets, input_lengths, target_lengths,
                                     lse, best, ctc_part, soft_part, dist, tlenc);
  finalize_kernel<<<1, 32, 0, stream>>>(ctc_part, soft_part, dist, tlenc, out);
}